// GraphConvOp_33346126086621
// MI455X (gfx1250) — compile-verified
//
#include <hip/hip_runtime.h>

// Problem constants from the reference (B=2, T=16, V=10000, F=32).
#define V_DIM    10000
#define F_DIM    32
#define BT_DIM   32          // B*T
#define GROUP_R  16          // output rows per WMMA tile group
#define N_GROUPS (V_DIM / GROUP_R)   // 625
#define CHUNK    1024        // edges staged in LDS per iteration
#define PAD      8           // sentinel edges appended after each chunk
#define WAVES_PB 8           // 256 threads / wave32

typedef __attribute__((ext_vector_type(2))) float        v2f;
typedef __attribute__((ext_vector_type(8))) float        v8f;
typedef __attribute__((ext_vector_type(4))) unsigned int u32x4;
typedef __attribute__((ext_vector_type(8))) int          i32x8;
typedef __attribute__((ext_vector_type(4))) int          i32x4;

// ---------------------------------------------------------------------------
// Kernel 1: row_ptr[r] = lower_bound(L_rows, r), r = 0..V  (rows are sorted)
// ---------------------------------------------------------------------------
__global__ void gcn_build_row_ptr(const int* __restrict__ rows, int nnz,
                                  int* __restrict__ row_ptr) {
    int r = blockIdx.x * blockDim.x + threadIdx.x;
    if (r > V_DIM) return;
    int lo = 0, hi = nnz;
    while (lo < hi) {
        int mid = (lo + hi) >> 1;
        if (rows[mid] < r) lo = mid + 1; else hi = mid;
    }
    row_ptr[r] = lo;
}

// ---------------------------------------------------------------------------
// LDS offset of a statically allocated __shared__ object (addrspacecast
// generic -> local, then ptrtoint gives the byte offset in the LDS segment).
// ---------------------------------------------------------------------------
__device__ inline unsigned gcn_lds_off(const void* p) {
    return (unsigned)(unsigned long long)
        (const __attribute__((address_space(3))) void*)p;
}

// ---------------------------------------------------------------------------
// TDM: DMA `ndw` dwords from global memory to LDS (1-row 2D tile descriptor).
// Emits tensor_load_to_lds (async-tensor path, tracked by TENSORcnt).
// ---------------------------------------------------------------------------
__device__ inline void gcn_tdm_copy_dwords(const void* gsrc, unsigned lds_addr,
                                           int ndw) {
    unsigned long long ga = (unsigned long long)gsrc;
    unsigned n = (unsigned)ndw;

    u32x4 g0;
    g0[0] = 1u;                                           // count=1, user mode
    g0[1] = lds_addr;                                     // lds_addr [63:32]
    g0[2] = (unsigned)(ga & 0xffffffffu);                 // global_addr lo
    g0[3] = (unsigned)((ga >> 32) & 0x1ffffffu)           // global_addr hi
          | (2u << 30);                                   // type = 2 (image)

    i32x8 g1;
    g1[0] = (int)(2u << 16);                              // data_size=4B, mask=0
    g1[1] = (int)((n & 0xffffu) << 16);                   // tensor_dim0[15:0]
    g1[2] = (int)(((n >> 16) & 0xffffu) | (1u << 16));    // dim0 hi | tensor_dim1=1
    g1[3] = (int)((n & 0xffffu) << 16);                   // tile_dim0 = n (n<=65535)
    g1[4] = 1;                                            // tile_dim1=1, tile_dim2=0
    g1[5] = (int)n;                                       // tensor_dim0_stride lo
    g1[6] = 0;
    g1[7] = 0;

    i32x4 z4 = {0, 0, 0, 0};                              // groups 2/3 unused (2D)
    i32x8 z8 = {0, 0, 0, 0, 0, 0, 0, 0};                  // trailing group unused
    __builtin_amdgcn_tensor_load_to_lds(g0, g1, z4, z4, z8, 0);
}

// ---------------------------------------------------------------------------
// Kernel 2: sorted-COO segment-sum SpMM via V_WMMA_F32_16X16X4_F32.
// One wave handles (row-group g, bt slice) and BOTH 16-wide feature tiles,
// reusing the A matrix (vals masked by row match) for two WMMAs per step.
// All 8 waves of a block share g -> edge tuples staged once via TDM.
// ---------------------------------------------------------------------------
__global__ void __launch_bounds__(256)
gcn_spmm_wmma(const float* __restrict__ x,      // [BT, V, F] fp32
              const float* __restrict__ vals,
              const int*   __restrict__ rows,
              const int*   __restrict__ cols,
              const int*   __restrict__ row_ptr,
              float* __restrict__ out) {
    __shared__ float s_vals[CHUNK + PAD];
    __shared__ int   s_rows[CHUNK + PAD];
    __shared__ int   s_cols[CHUNK + PAD];

    const int lane = threadIdx.x & 31;
    const int wave = threadIdx.x >> 5;
    const int w    = blockIdx.x * WAVES_PB + wave;
    const int bt   = w & (BT_DIM - 1);
    const int g    = w >> 5;                  // uniform across the block (8 | 32)

    const int rowbase = g * GROUP_R;
    const int half    = lane >> 4;            // K-half per ISA VGPR layout
    const int idx     = lane & 15;            // M (for A) / N (for B)
    const int k0      = half * 2;
    const int rid     = rowbase + idx;        // row this lane matches in A

    const float*   xbt = x + (size_t)bt * (V_DIM * F_DIM);
    const unsigned f0  = (unsigned)idx;       // feature tile 0 column
    const unsigned f1  = (unsigned)(16 + idx);// feature tile 1 column

    int start = __builtin_amdgcn_readfirstlane(row_ptr[rowbase]);
    int end   = __builtin_amdgcn_readfirstlane(row_ptr[rowbase + GROUP_R]);

    v8f acc0 = {};   // C/D tile: 16 rows x features [0,16)
    v8f acc1 = {};   // C/D tile: 16 rows x features [16,32)

    for (int base = start; base < end; base += CHUNK) {
        int len = end - base;
        if (len > CHUNK) len = CHUNK;

        if (wave == 0) {
            gcn_tdm_copy_dwords(vals + base, gcn_lds_off(s_vals), len);
            gcn_tdm_copy_dwords(rows + base, gcn_lds_off(s_rows), len);
            gcn_tdm_copy_dwords(cols + base, gcn_lds_off(s_cols), len);
            __builtin_amdgcn_s_wait_tensorcnt(0);
            if (lane < PAD) {                 // sentinel edges: contribute zero
                s_rows[len + lane] = -1;
                s_vals[len + lane] = 0.0f;
                s_cols[len + lane] = 0;
            }
        }
        __syncthreads();

        // Unrolled x2: 8 edges, 4 WMMAs per iteration. No guards needed —
        // sentinel pads give A==0 so out-of-range steps contribute nothing.
        for (int s = 0; s < len; s += 8) {
#pragma unroll
            for (int u = 0; u < 2; ++u) {
                const int e0 = s + 4 * u + k0;
                const int e1 = e0 + 1;
                const int      r0 = s_rows[e0], r1 = s_rows[e1];
                const float    v0 = s_vals[e0], v1 = s_vals[e1];
                const unsigned c0 = (unsigned)s_cols[e0] << 5;
                const unsigned c1 = (unsigned)s_cols[e1] << 5;

                v2f A;                        // A[m=idx, k=k0 / k0+1]
                A.x = (r0 == rid) ? v0 : 0.0f;
                A.y = (r1 == rid) ? v1 : 0.0f;

                v2f B0, B1;                   // B[k, n=idx] = x[col_k, n]
                B0.x = xbt[c0 + f0];
                B0.y = xbt[c1 + f0];
                B1.x = xbt[c0 + f1];
                B1.y = xbt[c1 + f1];

                acc0 = __builtin_amdgcn_wmma_f32_16x16x4_f32(
                    false, A, false, B0, (short)0, acc0, false, false);
                acc1 = __builtin_amdgcn_wmma_f32_16x16x4_f32(
                    false, A, false, B1, (short)0, acc1, false, false);
            }
        }
        __syncthreads();   // protect LDS before the next chunk overwrites it
    }

    // Store C tiles: VGPR j holds M = half*8 + j, N = idx.
    // Every (bt,v,f) is written exactly once across the grid.
    float* obt = out + (size_t)bt * (V_DIM * F_DIM);
    const int mbase = rowbase + half * 8;
#pragma unroll
    for (int j = 0; j < 8; ++j) {
        float* orow = obt + (size_t)(mbase + j) * F_DIM;
        orow[f0] = acc0[j];
        orow[f1] = acc1[j];
    }
}

// ---------------------------------------------------------------------------
extern "C" void kernel_launch(void* const* d_in, const int* in_sizes, int n_in,
                              void* d_out, int out_size, void* d_ws, size_t ws_size,
                              hipStream_t stream) {
    const float* x    = (const float*)d_in[0];   // inputs [B,T,V,F]
    const float* vals = (const float*)d_in[1];   // L_vals
    const int*   rows = (const int*)d_in[2];     // L_rows (sorted)
    const int*   cols = (const int*)d_in[3];     // L_cols
    const int    nnz  = in_sizes[1];

    int* row_ptr = (int*)d_ws;                   // V+1 ints of scratch

    gcn_build_row_ptr<<<(V_DIM + 1 + 255) / 256, 256, 0, stream>>>(rows, nnz, row_ptr);

    const int total_waves = N_GROUPS * BT_DIM;             // 20000
    const int blocks      = total_waves / WAVES_PB;        // 2500
    gcn_spmm_wmma<<<blocks, 256, 0, stream>>>(x, vals, rows, cols, row_ptr,
                                              (float*)d_out);
}